// FarthestSubsample_2765958938835
// MI455X (gfx1250) — compile-verified
//
#include <hip/hip_runtime.h>
#include <stdint.h>

typedef __attribute__((ext_vector_type(2))) float v2f;
typedef __attribute__((ext_vector_type(8))) float v8f;

constexpr int NB = 32;      // batches
constexpr int NN = 4096;    // points
constexpr int ND = 128;     // value channels
constexpr int NP = 2048;    // samples
constexpr int THREADS = 1024;               // 32 waves (wave32)
constexpr int WAVES = THREADS / 32;
constexpr int PTS_PER_WAVE = NN / WAVES;    // 128
constexpr int TILES = PTS_PER_WAVE / 16;    // 8 WMMA tiles per wave

struct __align__(16) FpsSmem {
  float coords[3 * NN];                  // staged copy of coords[b]: [3][NN]
  unsigned long long red[2][WAVES];      // double-buffered per-wave argmax partials
};

// ---------------- wave32 u64-max butterfly (VALU permlanes, no LDS) ----------
#if __has_builtin(__builtin_amdgcn_permlane16) && __has_builtin(__builtin_amdgcn_permlanex16)
#define FPS_USE_PERMLANE 1
__device__ inline unsigned long long pl16_xor_u64(unsigned long long v,
                                                  unsigned s0, unsigned s1) {
  unsigned lo = (unsigned)v, hi = (unsigned)(v >> 32);
  lo = (unsigned)__builtin_amdgcn_permlane16(lo, lo, s0, s1, false, false);
  hi = (unsigned)__builtin_amdgcn_permlane16(hi, hi, s0, s1, false, false);
  return ((unsigned long long)hi << 32) | (unsigned long long)lo;
}
__device__ inline unsigned long long plx16_u64(unsigned long long v) {
  unsigned lo = (unsigned)v, hi = (unsigned)(v >> 32);
  lo = (unsigned)__builtin_amdgcn_permlanex16(lo, lo, 0x76543210u, 0xFEDCBA98u, false, false);
  hi = (unsigned)__builtin_amdgcn_permlanex16(hi, hi, 0x76543210u, 0xFEDCBA98u, false, false);
  return ((unsigned long long)hi << 32) | (unsigned long long)lo;
}
__device__ inline unsigned long long wave_max_u64(unsigned long long k) {
  unsigned long long o;
  o = pl16_xor_u64(k, 0x67452301u, 0xEFCDAB89u); k = (o > k) ? o : k;  // xor 1
  o = pl16_xor_u64(k, 0x54761032u, 0xDCFE98BAu); k = (o > k) ? o : k;  // xor 2
  o = pl16_xor_u64(k, 0x32107654u, 0xBA98FEDCu); k = (o > k) ? o : k;  // xor 4
  o = pl16_xor_u64(k, 0xFEDCBA98u, 0x76543210u); k = (o > k) ? o : k;  // xor 8
  o = plx16_u64(k);                              k = (o > k) ? o : k;  // xor 16
  return k;
}
#else
__device__ inline unsigned long long shflx_u64(unsigned long long v, int m) {
  unsigned lo = (unsigned)__shfl_xor((int)(unsigned)(v & 0xffffffffull), m, 32);
  unsigned hi = (unsigned)__shfl_xor((int)(unsigned)(v >> 32), m, 32);
  return ((unsigned long long)hi << 32) | (unsigned long long)lo;
}
__device__ inline unsigned long long wave_max_u64(unsigned long long k) {
#pragma unroll
  for (int m = 16; m >= 1; m >>= 1) {
    unsigned long long o = shflx_u64(k, m);
    k = (o > k) ? o : k;
  }
  return k;
}
#endif

__global__ __launch_bounds__(THREADS)
void fps_kernel(const float* __restrict__ coords,
                int* __restrict__ fps_idx,
                float* __restrict__ out_coords) {
  __shared__ FpsSmem sm;
  const int b = blockIdx.x;
  const int tid = (int)threadIdx.x;
  const int w = tid >> 5;
  const int lane = tid & 31;
  const int ln = lane & 15;      // column / point-in-tile
  const int h = lane >> 4;       // half-wave: selects K rows {0,1} vs {2,3}
  const int wbase = w * PTS_PER_WAVE;

  // ---- Stage coords[b] (48KB contiguous) into LDS via CDNA5 async copy ----
  const float* cbase = coords + (size_t)b * 3 * NN;
  for (int k = tid; k < (3 * NN) / 4; k += THREADS) {
    unsigned int ldsoff = (unsigned int)(uintptr_t)(&sm.coords[4 * k]);
    unsigned long long ga = (unsigned long long)(uintptr_t)(cbase + 4 * k);
    asm volatile("global_load_async_to_lds_b128 %0, %1, off"
                 :: "v"(ldsoff), "v"(ga) : "memory");
  }
  asm volatile("s_wait_asynccnt 0x0" ::: "memory");
  __syncthreads();

  // ---- Build static B tiles: B[4][16] per tile = (x,y,z,|p|^2) of 16 points.
  // Layout (mirror of documented A 16x4): v0 = {K0 | K2}, v1 = {K1 | K3}.
  v2f btile[TILES];
  unsigned int kidx[TILES];      // precomputed (0xffffffff - point_index)
#pragma unroll
  for (int t = 0; t < TILES; ++t) {
    int p = wbase + t * 16 + ln;
    float x = sm.coords[p];
    float y = sm.coords[NN + p];
    float z = sm.coords[2 * NN + p];
    float sq = x * x + y * y + z * z;
    btile[t][0] = h ? z : x;
    btile[t][1] = h ? sq : y;
    kidx[t] = 0xffffffffu - (unsigned int)p;
  }

  // Running min distances kept as raw IEEE bits: non-negative floats order
  // identically as u32, so the update is a single v_min_u32 (no fminf
  // canonicalization), and the sign bit can never leak into the argmax key.
  unsigned int runs[TILES];
#pragma unroll
  for (int t = 0; t < TILES; ++t) runs[t] = __float_as_uint(1e10f);

  // Loop-carried state lives in registers (uniform across threads).
  unsigned int curFar = 0u;
  float cx = sm.coords[0], cy = sm.coords[NN], cz = sm.coords[2 * NN];
  float csq = cx * cx + cy * cy + cz * cz;

  for (int i = 0; i < NP; ++i) {
    // Emit BEFORE update (matches reference loop semantics).
    if (tid == 0) {
      fps_idx[b * NP + i] = (int)curFar;
      out_coords[(size_t)b * 3 * NP + i] = cx;
      out_coords[(size_t)b * 3 * NP + NP + i] = cy;
      out_coords[(size_t)b * 3 * NP + 2 * NP + i] = cz;
    }

    // A = 16 replicated rows of (-2cx, -2cy, -2cz, 1); C = |c|^2 broadcast.
    // D[m][n] = |p_n|^2 - 2 p_n.c + |c|^2 = ||p_n - c||^2  (rows replicated).
    v2f a;
    a[0] = h ? (-2.0f * cz) : (-2.0f * cx);
    a[1] = h ? 1.0f : (-2.0f * cy);
    v8f cin = {csq, csq, csq, csq, csq, csq, csq, csq};

    unsigned long long best = 0ull;
#pragma unroll
    for (int t = 0; t < TILES; ++t) {
      v8f d = __builtin_amdgcn_wmma_f32_16x16x4_f32(
          false, a, false, btile[t], (short)0, cin, false, false);
      unsigned int db = __float_as_uint(d[0]);     // point = lane&15 of tile t
      unsigned int nd = (db < runs[t]) ? db : runs[t];   // v_min_u32
      runs[t] = nd;
      // key: larger distance wins; ties -> smaller index (first-max, like argmax)
      unsigned long long key =
          ((unsigned long long)nd << 32) | (unsigned long long)kidx[t];
      best = (key > best) ? key : best;
    }

    // Stage 1: wave32 max (pure VALU permlane butterfly).
    best = wave_max_u64(best);
    if (lane == 0) sm.red[i & 1][w] = best;
    __syncthreads();   // the only barrier per iteration (red[] is double-buffered)

    // Stage 2: every wave redundantly reduces the 32 partials -> all threads
    // hold the winner; decode + refetch centroid without a second barrier.
    unsigned long long k2 = sm.red[i & 1][lane];
    k2 = wave_max_u64(k2);
    unsigned int nf = 0xffffffffu - (unsigned int)(k2 & 0xffffffffull);
    curFar = nf;
    cx = sm.coords[nf];
    cy = sm.coords[NN + nf];
    cz = sm.coords[2 * NN + nf];
    csq = cx * cx + cy * cy + cz * cz;
  }
}

__global__ __launch_bounds__(256)
void gather_values_kernel(const float* __restrict__ values,
                          const int* __restrict__ fps_idx,
                          float* __restrict__ out_values) {
  const int bd = (int)blockIdx.x;        // b*ND + d
  const int b = bd >> 7;                 // / ND
  const float* row = values + (size_t)bd * NN;
  const int* ib = fps_idx + b * NP;
  float* orow = out_values + (size_t)bd * NP;
  for (int s = (int)threadIdx.x; s < NP; s += 256) {
    orow[s] = row[ib[s]];                // gathered read (L2-resident), coalesced write
  }
}

extern "C" void kernel_launch(void* const* d_in, const int* in_sizes, int n_in,
                              void* d_out, int out_size, void* d_ws, size_t ws_size,
                              hipStream_t stream) {
  const float* coords = (const float*)d_in[0];   // [32, 3, 4096]
  const float* values = (const float*)d_in[1];   // [32, 128, 4096]
  (void)in_sizes; (void)n_in; (void)out_size; (void)ws_size;

  int* fps_idx = (int*)d_ws;                                   // 32*2048 ints
  float* out_coords = (float*)d_out;                           // [32,3,2048]
  float* out_values = (float*)d_out + (size_t)NB * 3 * NP;     // [32,128,2048]

  fps_kernel<<<NB, THREADS, 0, stream>>>(coords, fps_idx, out_coords);
  gather_values_kernel<<<NB * ND, 256, 0, stream>>>(values, fps_idx, out_values);
}